// NSLoss_13589276525289
// MI455X (gfx1250) — compile-verified
//
#include <hip/hip_runtime.h>
#include <hip/hip_bf16.h>

typedef float v2f __attribute__((ext_vector_type(2)));
typedef float v8f __attribute__((ext_vector_type(8)));

#define NPTS   8192
#define BATCH  2
#define VOXEL  0.1f
#define CTILES (NPTS / 16)          // 512 column tiles per batch

// ---------------------------------------------------------------------------
// per (array, batch) component-wise min with NaN -> +inf masking
// blockIdx.x in [0,4): bit1 = array (0=preds,1=gts), bit0 = batch
// ---------------------------------------------------------------------------
__global__ __launch_bounds__(256) void nsl_gmin(const float* __restrict__ preds,
                                                const float* __restrict__ gts,
                                                float* __restrict__ gmin /*[2][2][3]*/) {
    __shared__ float s0[256], s1[256], s2[256];
    const int arr = blockIdx.x >> 1;
    const int b   = blockIdx.x & 1;
    const float* src = (arr ? gts : preds) + (size_t)b * NPTS * 3;

    const float INF = __builtin_inff();
    float m0 = INF, m1 = INF, m2 = INF;
    for (int i = threadIdx.x; i < NPTS; i += 256) {
        float c0 = src[3 * i + 0];
        float c1 = src[3 * i + 1];
        float c2 = src[3 * i + 2];
        if (c0 != c0) c0 = INF;   // isnan -> inf (masked_fill semantics)
        if (c1 != c1) c1 = INF;
        if (c2 != c2) c2 = INF;
        m0 = fminf(m0, c0);
        m1 = fminf(m1, c1);
        m2 = fminf(m2, c2);
    }
    const int tid = threadIdx.x;
    s0[tid] = m0; s1[tid] = m1; s2[tid] = m2;
    __syncthreads();
    for (int st = 128; st > 0; st >>= 1) {
        if (tid < st) {
            s0[tid] = fminf(s0[tid], s0[tid + st]);
            s1[tid] = fminf(s1[tid], s1[tid + st]);
            s2[tid] = fminf(s2[tid], s2[tid + st]);
        }
        __syncthreads();
    }
    if (tid == 0) {
        gmin[arr * 6 + b * 3 + 0] = s0[0];
        gmin[arr * 6 + b * 3 + 1] = s1[0];
        gmin[arr * 6 + b * 3 + 2] = s2[0];
    }
}

// ---------------------------------------------------------------------------
// voxelize both arrays: out = (float)(int)((v - gmin[b][c]) / VOXEL)
// ---------------------------------------------------------------------------
__global__ __launch_bounds__(256) void nsl_voxelize(const float* __restrict__ preds,
                                                    const float* __restrict__ gts,
                                                    const float* __restrict__ gmin,
                                                    float* __restrict__ pv,
                                                    float* __restrict__ gv) {
    const int per = BATCH * NPTS * 3;
    int idx = blockIdx.x * blockDim.x + threadIdx.x;
    if (idx >= 2 * per) return;
    const int arr = idx / per;
    const int r   = idx % per;
    const int b   = r / (NPTS * 3);
    const int c   = r % 3;
    const float* in  = arr ? gts : preds;
    float*       out = arr ? gv  : pv;
    const float v = in[r];
    const float m = gmin[arr * 6 + b * 3 + c];
    out[r] = (float)(int)((v - m) / VOXEL);   // trunc toward zero, like .int()
}

// ---------------------------------------------------------------------------
// Pre-pack the WMMA B operand for all four column roles.
// B 4x16 fp32 layout: lanes 0-15 hold K=0,1 in vgpr0,1; lanes 16-31 hold K=2,3.
// Packed col n: [y0, y1, y2, yy].  Storage: pk[role][b][ct][lane][2].
// ---------------------------------------------------------------------------
__global__ __launch_bounds__(256) void nsl_pack_cols(const float* __restrict__ gts,
                                                     const float* __restrict__ preds,
                                                     const float* __restrict__ gv,
                                                     const float* __restrict__ pv,
                                                     float* __restrict__ pk) {
    const int total = 4 * BATCH * CTILES * 32;
    int idx = blockIdx.x * blockDim.x + threadIdx.x;
    if (idx >= total) return;
    const int lane = idx & 31;
    const int ct   = (idx >> 5) & (CTILES - 1);
    const int b    = (idx >> 14) & (BATCH - 1);
    const int role = idx >> 15;

    const float* src = (role == 0) ? gts : (role == 1) ? preds : (role == 2) ? gv : pv;
    const float* yb  = src + (size_t)b * NPTS * 3;

    const int l16  = lane & 15;
    const int half = lane >> 4;
    const int n    = ct * 16 + l16;

    const float y0 = yb[3 * n + 0];
    const float y1 = yb[3 * n + 1];
    const float y2 = yb[3 * n + 2];
    const float yy = fmaf(y0, y0, fmaf(y1, y1, y2 * y2));

    pk[(size_t)idx * 2 + 0] = half ? y2 : y0;
    pk[(size_t)idx * 2 + 1] = half ? yy : y1;
}

// ---------------------------------------------------------------------------
// lane-xor min via ds_swizzle (group-of-32 xor mode: offset = xor<<10 | and=0x1f)
// ---------------------------------------------------------------------------
template <int IMM>
__device__ __forceinline__ void swz_min(v8f& r) {
#pragma unroll
    for (int v = 0; v < 8; ++v) {
        int t = __builtin_amdgcn_ds_swizzle(__float_as_int(r[v]), IMM);
        r[v] = fminf(r[v], __int_as_float(t));
    }
}

// ---------------------------------------------------------------------------
// One-sided chamfer with V_WMMA_F32_16X16X4_F32:
//   A row m = [-2x0, -2x1, -2x2, 1], B col n = [y0, y1, y2, yy],
//   C[m][n] = xx_m  =>  D[m][n] = ||x_m - y_n||^2 exactly in fp32.
// Each wave owns 16 rows; inner loop = 1 coalesced b64 load + 1 WMMA + mins.
// Deterministic: per-wave partial sums, no float atomics.
// ---------------------------------------------------------------------------
__global__ __launch_bounds__(256) void nsl_chamfer_min(const float* __restrict__ X,
                                                       const float* __restrict__ Ypack,
                                                       float* __restrict__ partial) {
    const int lane = threadIdx.x & 31;
    const int wave = threadIdx.x >> 5;
    const int gw   = blockIdx.x * (blockDim.x >> 5) + wave;

    const int b       = gw / CTILES;
    const int rowBase = (gw % CTILES) * 16;
    const int l16     = lane & 15;
    const int half    = lane >> 4;

    const float* xb = X + (size_t)b * NPTS * 3;

    // ---- A tile (loop-invariant): lanes 0-15 carry K=0,1; lanes 16-31 K=2,3
    const int m = rowBase + l16;
    const float x0 = xb[3 * m + 0];
    const float x1 = xb[3 * m + 1];
    const float x2 = xb[3 * m + 2];
    const float xx = fmaf(x0, x0, fmaf(x1, x1, x2 * x2));

    v2f a;
    a.x = half ? (-2.0f * x2) : (-2.0f * x0);
    a.y = half ? 1.0f         : (-2.0f * x1);

    // ---- C tile (loop-invariant): C[m][n] = xx_m ; M = vgpr + 8*half
    v8f cmat;
#pragma unroll
    for (int v = 0; v < 8; ++v)
        cmat[v] = __shfl(xx, v + 8 * half, 32);

    v8f runmin;
#pragma unroll
    for (int v = 0; v < 8; ++v) runmin[v] = __builtin_inff();

    // ---- stream packed column tiles: b64 load + WMMA + min-accumulate
    const float* yp = Ypack + ((size_t)b * CTILES) * 64 + lane * 2;
    for (int ct = 0; ct < CTILES; ++ct) {
        const v2f bm = *reinterpret_cast<const v2f*>(yp + (size_t)ct * 64);
        v8f d = __builtin_amdgcn_wmma_f32_16x16x4_f32(
            /*neg_a=*/false, a, /*neg_b=*/false, bm,
            /*c_mod=*/(short)0, cmat, /*reuse_a=*/false, /*reuse_b=*/false);
#pragma unroll
        for (int v = 0; v < 8; ++v) runmin[v] = fminf(runmin[v], d[v]);
    }

    // ---- min over N: xor butterfly inside each 16-lane half (N = lane%16)
    swz_min<(1 << 10) | 0x1f>(runmin);   // SWAPX1
    swz_min<(2 << 10) | 0x1f>(runmin);   // SWAPX2
    swz_min<(4 << 10) | 0x1f>(runmin);   // SWAPX4
    swz_min<(8 << 10) | 0x1f>(runmin);   // SWAPX8

    // each half now holds the 8 row-mins for M = v + 8*half
    float s = 0.0f;
#pragma unroll
    for (int v = 0; v < 8; ++v) s += runmin[v];
    if (l16 == 0) partial[gw * 2 + half] = s;   // deterministic partial
}

// ---------------------------------------------------------------------------
// fixed-order final reduction: out = scale * sum(partial[0..n))
// ---------------------------------------------------------------------------
__global__ __launch_bounds__(256) void nsl_reduce(const float* __restrict__ partial,
                                                  float* __restrict__ out,
                                                  float scale, int n) {
    __shared__ float s[256];
    float acc = 0.0f;
    for (int i = threadIdx.x; i < n; i += 256) acc += partial[i];
    s[threadIdx.x] = acc;
    __syncthreads();
    for (int st = 128; st > 0; st >>= 1) {
        if (threadIdx.x < st) s[threadIdx.x] += s[threadIdx.x + st];
        __syncthreads();
    }
    if (threadIdx.x == 0) out[0] = s[0] * scale;
}

// ---------------------------------------------------------------------------
extern "C" void kernel_launch(void* const* d_in, const int* in_sizes, int n_in,
                              void* d_out, int out_size, void* d_ws, size_t ws_size,
                              hipStream_t stream) {
    const float* preds = (const float*)d_in[0];
    const float* gts   = (const float*)d_in[1];
    float* out = (float*)d_out;

    // ws layout (floats)
    float* ws      = (float*)d_ws;
    float* gmin    = ws;                                   // 12 (pad 16)
    float* pv      = ws + 16;                              // 2*8192*3
    float* gv      = pv + (size_t)BATCH * NPTS * 3;        // 2*8192*3
    float* pk      = gv + (size_t)BATCH * NPTS * 3;        // 4*2*512*64 = 262144
    float* partial = pk + (size_t)4 * BATCH * CTILES * 64; // 4*2048 = 8192

    nsl_gmin<<<4, 256, 0, stream>>>(preds, gts, gmin);

    const int nel = 2 * BATCH * NPTS * 3;
    nsl_voxelize<<<(nel + 255) / 256, 256, 0, stream>>>(preds, gts, gmin, pv, gv);

    const int npack = 4 * BATCH * CTILES * 32;
    nsl_pack_cols<<<(npack + 255) / 256, 256, 0, stream>>>(gts, preds, gv, pv, pk);

    const size_t roleStride = (size_t)BATCH * CTILES * 64; // floats per packed role
    dim3 grid(BATCH * CTILES / 8);                         // 8 waves/block, wave32
    dim3 block(256);

    // pass p: rows X vs packed cols (role p); partial block of 2048 per pass
    nsl_chamfer_min<<<grid, block, 0, stream>>>(preds, pk + 0 * roleStride, partial + 0 * 2048); // d1 : preds vs gts
    nsl_chamfer_min<<<grid, block, 0, stream>>>(gts,   pk + 1 * roleStride, partial + 1 * 2048); // d2 : gts  vs preds
    nsl_chamfer_min<<<grid, block, 0, stream>>>(pv,    pk + 2 * roleStride, partial + 2 * 2048); // d1v: pv   vs gv
    nsl_chamfer_min<<<grid, block, 0, stream>>>(gv,    pk + 3 * roleStride, partial + 3 * 2048); // d2v: gv   vs pv

    const float scale = 1.0f / (float)(BATCH * NPTS);      // mean over [B,N]
    nsl_reduce<<<1, 256, 0, stream>>>(partial, out, scale, 4 * 2048);
}